// Generator_42795054137896
// MI455X (gfx1250) — compile-verified
//
#include <hip/hip_runtime.h>
#include <hip/hip_bf16.h>

// ---------------------------------------------------------------------------
// Tiny encoder-decoder transformer for MI455X (gfx1250).
//   B=64, S=64, U=16, EMB=10, D=34 (2 heads, dh=17), DD=3, FF=4, NL=4.
//  * embed+posenc kernel  (elementwise)
//  * encoder kernel: 1 block per batch element, 256 thr (8 waves),
//    activations + transposed/padded weights in LDS, all matmuls via
//    V_WMMA_F32_16X16X4_F32 (fp32 WMMA) with guard-free LDS fragments
//  * decoder kernel: 1 block per batch element, exact KV-cached
//    incremental autoregressive decoding (63 steps)
// ---------------------------------------------------------------------------

typedef float v2f __attribute__((ext_vector_type(2)));
typedef float v8f __attribute__((ext_vector_type(8)));

static __device__ __forceinline__ v8f wmma4(v2f a, v2f b, v8f c) {
  // D = A(16x4,f32) * B(4x16,f32) + C(16x16,f32), full fp32 WMMA
  return __builtin_amdgcn_wmma_f32_16x16x4_f32(false, a, false, b, (short)0, c,
                                               false, false);
}

// ------------------------------ embed ---------------------------------------
__global__ __launch_bounds__(256) void embed_kernel(
    const float* __restrict__ src, const float* __restrict__ wemb,
    const float* __restrict__ semb, float* __restrict__ xg) {
  int idx = blockIdx.x * blockDim.x + threadIdx.x;  // over B*S = 4096
  if (idx >= 64 * 64) return;
  int b = idx >> 6;
  const float* sp = src + idx * 16;
  float* xp = xg + idx * 36;  // stride 36 = K padded, cols 34..35 = 0
  int widx = (int)sp[14]; widx = widx < 0 ? 0 : (widx > 9999 ? 9999 : widx);
  int sidx = (int)sp[15]; sidx = sidx < 0 ? 0 : (sidx > 399 ? 399 : sidx);
  float v[36];
  for (int c = 0; c < 14; ++c) v[c] = sp[c];
  for (int c = 0; c < 10; ++c) v[14 + c] = wemb[widx * 10 + c];
  for (int c = 0; c < 10; ++c) v[24 + c] = semb[sidx * 10 + c];
  // positional-encoding quirk: pe row b added to every sequence position of b
  for (int i = 0; i < 17; ++i) {
    float div = expf((float)(2 * i) * (-9.210340371976184f / 34.0f));
    float ang = (float)b * div;
    v[2 * i]     += sinf(ang);
    v[2 * i + 1] += cosf(ang);
  }
  v[34] = 0.f; v[35] = 0.f;
  for (int c = 0; c < 36; ++c) xp[c] = v[c];
}

// ------------------------------ encoder -------------------------------------
// Y layout (stride 114), zero-padded per head so WMMA fragments need no guards:
//   Q0: 0..19 (17 used)   Q1: 20..39 (17 used)
//   K0: 40..59 (17 used)  K1: 60..79 (17 used)
//   V0: 80..96            V1: 97..113
// SCRATCH (4096 floats) phases:
//   A: qkv weight^T  [36][112]  (4032)
//   B: attention scores [64][64]
//   C: Z [64][36] (0..2303) + out weight^T [36][48] (2304..4031)
//   D: FF hidden [64][4] (0..255)
__global__ __launch_bounds__(256) void encoder_kernel(
    const float* __restrict__ xg,
    const float* __restrict__ qkv_w, const float* __restrict__ qkv_b,
    const float* __restrict__ out_w, const float* __restrict__ out_b,
    const float* __restrict__ ln1_w, const float* __restrict__ ln1_b,
    const float* __restrict__ lin1_w, const float* __restrict__ lin1_b,
    const float* __restrict__ lin2_w, const float* __restrict__ lin2_b,
    const float* __restrict__ ln2_w, const float* __restrict__ ln2_b,
    const float* __restrict__ norm_w, const float* __restrict__ norm_b,
    const float* __restrict__ eo_w, const float* __restrict__ eo_b,
    float* __restrict__ memg) {
  __shared__ float X[64][36];     // activations, cols 34..35 stay zero
  __shared__ float Y[64][114];    // padded q/k/v (see above)
  __shared__ float SCRATCH[4096]; // multi-phase scratch
  __shared__ float T[64][36];     // attn output / ff output, cols 34..35 zero

  const int tid = threadIdx.x;
  const int w = tid >> 5;        // wave id 0..7
  const int lane = tid & 31;
  const int lr = lane & 15;      // row/col within half-wave
  const int lh = lane >> 4;      // half-wave select
  const int b = blockIdx.x;

  for (int idx = tid; idx < 64 * 36; idx += 256) {
    ((float*)X)[idx] = xg[b * 64 * 36 + idx];
    ((float*)T)[idx] = 0.f;
  }
  for (int idx = tid; idx < 64 * 114; idx += 256) ((float*)Y)[idx] = 0.f;
  __syncthreads();

  for (int l = 0; l < 4; ++l) {
    // ---- phase A: stage qkv weights transposed+padded into SCRATCH ----
    {
      const float* Wp = qkv_w + l * 102 * 34;
      for (int idx = tid; idx < 4032; idx += 256) SCRATCH[idx] = 0.f;
      __syncthreads();
      for (int idx = tid; idx < 102 * 34; idx += 256) {
        int n = idx / 34, k = idx % 34;       // coalesced global read
        SCRATCH[k * 112 + n] = Wp[idx];
      }
    }
    __syncthreads();

    // ---- QKV projection: Y = X[64,34] @ W^T + b  (WMMA, guard-free) ----
    {
      const float* Bp = qkv_b + l * 102;
      for (int t = w; t < 28; t += 8) {  // 4 x 7 tiles
        int mt = t / 7, nt = t % 7;
        int c = nt * 16 + lr;            // qkv channel 0..111
        v8f acc = {};
        for (int kc = 0; kc < 9; ++kc) {  // K=34 padded to 36
          int k0 = kc * 4 + lh * 2;
          v2f a, bb;
          a.x = X[mt * 16 + lr][k0];
          a.y = X[mt * 16 + lr][k0 + 1];
          bb.x = SCRATCH[k0 * 112 + c];
          bb.y = SCRATCH[(k0 + 1) * 112 + c];
          acc = wmma4(a, bb, acc);
        }
        if (c < 102) {
          // branch-free remap into per-head padded layout:
          //   q: h*20+ch, k: 40+h*20+ch, v: 80+ch17
          int col = c + 3 * ((c >= 17) + (c >= 34) + (c >= 51) + (c >= 68));
          float bias = Bp[c];
          for (int j = 0; j < 8; ++j)
            Y[mt * 16 + lh * 8 + j][col] = acc[j] + bias;
        }
      }
    }
    __syncthreads();

    // ---- phase B: attention, one head at a time (dh = 17) ----
    for (int h = 0; h < 2; ++h) {
      const int qoff = h * 20, koff = 40 + h * 20, voff = 80 + h * 17;
      const float scale = 0.24253562503633297f;  // 1/sqrt(17)
      for (int t = w; t < 16; t += 8) {          // scores: 4x4 tiles
        int mt = t >> 2, nt = t & 3;
        v8f acc = {};
        for (int kc = 0; kc < 5; ++kc) {         // K=17 padded to 20, pads zero
          int k0 = kc * 4 + lh * 2;
          v2f a, bb;
          a.x  = Y[mt * 16 + lr][qoff + k0];
          a.y  = Y[mt * 16 + lr][qoff + k0 + 1];
          bb.x = Y[nt * 16 + lr][koff + k0];
          bb.y = Y[nt * 16 + lr][koff + k0 + 1];
          acc = wmma4(a, bb, acc);
        }
        for (int j = 0; j < 8; ++j)
          SCRATCH[(mt * 16 + lh * 8 + j) * 64 + nt * 16 + lr] = acc[j] * scale;
      }
      __syncthreads();
      if (tid < 64) {  // row softmax
        float m = -1e30f;
        for (int k = 0; k < 64; ++k) m = fmaxf(m, SCRATCH[tid * 64 + k]);
        float s = 0.f;
        for (int k = 0; k < 64; ++k) {
          float e = __expf(SCRATCH[tid * 64 + k] - m);
          SCRATCH[tid * 64 + k] = e; s += e;
        }
        float inv = 1.f / s;
        for (int k = 0; k < 64; ++k) SCRATCH[tid * 64 + k] *= inv;
      }
      __syncthreads();
      {                                   // attn@V: 4x2 tiles (1 per wave)
        int t = w;
        int mt = t >> 1, nt = t & 1;
        int n0 = nt * 16 + lr;
        int vn = n0 < 17 ? n0 : 16;       // clamp: cols >=17 are discarded
        v8f acc = {};
        for (int kc = 0; kc < 16; ++kc) {  // K = 64
          int k0 = kc * 4 + lh * 2;
          v2f a, bb;
          a.x = SCRATCH[(mt * 16 + lr) * 64 + k0];
          a.y = SCRATCH[(mt * 16 + lr) * 64 + k0 + 1];
          bb.x = Y[k0    ][voff + vn];
          bb.y = Y[k0 + 1][voff + vn];
          acc = wmma4(a, bb, acc);
        }
        if (n0 < 17)
          for (int j = 0; j < 8; ++j)
            T[mt * 16 + lh * 8 + j][h * 17 + n0] = acc[j];
      }
      __syncthreads();
    }

    // ---- phase C: stage out_w^T, then output projection ----
    float* Zf    = SCRATCH;          // [64][36]
    float* outWT = SCRATCH + 2304;   // [36][48]
    {
      const float* Ow = out_w + l * 34 * 34;
      for (int idx = tid; idx < 36 * 48; idx += 256) outWT[idx] = 0.f;
      __syncthreads();
      for (int idx = tid; idx < 34 * 34; idx += 256) {
        int n = idx / 34, k = idx % 34;
        outWT[k * 48 + n] = Ow[idx];
      }
    }
    __syncthreads();
    {
      const float* Ob = out_b + l * 34;
      for (int t = w; t < 12; t += 8) {
        int mt = t / 3, nt = t % 3;
        int n = nt * 16 + lr;
        v8f acc = {};
        for (int kc = 0; kc < 9; ++kc) {
          int k0 = kc * 4 + lh * 2;
          v2f a, bb;
          a.x = T[mt * 16 + lr][k0];
          a.y = T[mt * 16 + lr][k0 + 1];
          bb.x = outWT[k0 * 48 + n];
          bb.y = outWT[(k0 + 1) * 48 + n];
          acc = wmma4(a, bb, acc);
        }
        if (n < 34) {
          float bias = Ob[n];
          for (int j = 0; j < 8; ++j)
            Zf[(mt * 16 + lh * 8 + j) * 36 + n] = acc[j] + bias;
        }
      }
    }
    __syncthreads();

    // ---- residual + LN1 ----
    if (tid < 64) {
      const float* wv = ln1_w + l * 34;
      const float* bv = ln1_b + l * 34;
      float v[34], mu = 0.f;
      for (int c = 0; c < 34; ++c) { v[c] = X[tid][c] + Zf[tid * 36 + c]; mu += v[c]; }
      mu *= (1.f / 34.f);
      float var = 0.f;
      for (int c = 0; c < 34; ++c) { float d = v[c] - mu; var += d * d; }
      var *= (1.f / 34.f);
      float inv = rsqrtf(var + 1e-5f);
      for (int c = 0; c < 34; ++c) X[tid][c] = (v[c] - mu) * inv * wv[c] + bv[c];
    }
    __syncthreads();

    // ---- phase D: feed-forward (34 -> 4 -> 34), scalar ----
    float* Hf = SCRATCH;  // [64][4]
    {
      int row = tid >> 2, cf = tid & 3;
      const float* w1 = lin1_w + (l * 4 + cf) * 34;
      float s = lin1_b[l * 4 + cf];
      for (int k = 0; k < 34; ++k) s += X[row][k] * w1[k];
      Hf[row * 4 + cf] = fmaxf(s, 0.f);
    }
    __syncthreads();
    for (int idx = tid; idx < 64 * 34; idx += 256) {
      int row = idx / 34, c = idx % 34;
      const float* w2 = lin2_w + (l * 34 + c) * 4;
      float s = lin2_b[l * 34 + c];
      for (int f = 0; f < 4; ++f) s += Hf[row * 4 + f] * w2[f];
      T[row][c] = s;
    }
    __syncthreads();
    if (tid < 64) {  // residual + LN2
      const float* wv = ln2_w + l * 34;
      const float* bv = ln2_b + l * 34;
      float v[34], mu = 0.f;
      for (int c = 0; c < 34; ++c) { v[c] = X[tid][c] + T[tid][c]; mu += v[c]; }
      mu *= (1.f / 34.f);
      float var = 0.f;
      for (int c = 0; c < 34; ++c) { float d = v[c] - mu; var += d * d; }
      var *= (1.f / 34.f);
      float inv = rsqrtf(var + 1e-5f);
      for (int c = 0; c < 34; ++c) X[tid][c] = (v[c] - mu) * inv * wv[c] + bv[c];
    }
    __syncthreads();
  }

  // ---- final LayerNorm + memory projection to DD=3 ----
  if (tid < 64) {
    float v[34], mu = 0.f;
    for (int c = 0; c < 34; ++c) { v[c] = X[tid][c]; mu += v[c]; }
    mu *= (1.f / 34.f);
    float var = 0.f;
    for (int c = 0; c < 34; ++c) { float d = v[c] - mu; var += d * d; }
    var *= (1.f / 34.f);
    float inv = rsqrtf(var + 1e-5f);
    for (int c = 0; c < 34; ++c) X[tid][c] = (v[c] - mu) * inv * norm_w[c] + norm_b[c];
  }
  __syncthreads();
  for (int idx = tid; idx < 64 * 3; idx += 256) {
    int row = idx / 3, o = idx % 3;
    float s = eo_b[o];
    for (int k = 0; k < 34; ++k) s += X[row][k] * eo_w[o * 34 + k];
    memg[(b * 64 + row) * 3 + o] = s;
  }
}

// ------------------------------ decoder -------------------------------------
__global__ __launch_bounds__(64) void decoder_kernel(
    const float* __restrict__ memg, const float* __restrict__ src,
    const float* __restrict__ sa_qkv_w, const float* __restrict__ sa_qkv_b,
    const float* __restrict__ sa_out_w, const float* __restrict__ sa_out_b,
    const float* __restrict__ dln1_w, const float* __restrict__ dln1_b,
    const float* __restrict__ ca_qkv_w, const float* __restrict__ ca_qkv_b,
    const float* __restrict__ ca_out_w, const float* __restrict__ ca_out_b,
    const float* __restrict__ dln2_w, const float* __restrict__ dln2_b,
    const float* __restrict__ dlin1_w, const float* __restrict__ dlin1_b,
    const float* __restrict__ dlin2_w, const float* __restrict__ dlin2_b,
    const float* __restrict__ dln3_w, const float* __restrict__ dln3_b,
    float* __restrict__ out) {
  __shared__ float mem[64][3];
  __shared__ float kc[4][64][3], vc[4][64][3];   // cross-attn KV (precomputed)
  __shared__ float ksa[4][64][3], vsa[4][64][3]; // self-attn KV cache
  __shared__ float tgt[64][3];
  __shared__ float yv[3], qv[3];
  __shared__ float sc[64];

  const int b = blockIdx.x;
  const int j = threadIdx.x;  // 0..63
  const float isq3 = 0.5773502691896258f;  // 1/sqrt(3)

  for (int c = 0; c < 3; ++c) {
    mem[j][c] = memg[(b * 64 + j) * 3 + c];
    tgt[j][c] = 0.f;
  }
  __syncthreads();
  // precompute cross-attention K/V for each layer
  for (int l = 0; l < 4; ++l) {
    const float* W = ca_qkv_w + l * 27;
    const float* Bv = ca_qkv_b + l * 9;
    for (int c = 0; c < 3; ++c) {
      kc[l][j][c] = Bv[3 + c] + mem[j][0] * W[(3 + c) * 3 + 0] +
                    mem[j][1] * W[(3 + c) * 3 + 1] + mem[j][2] * W[(3 + c) * 3 + 2];
      vc[l][j][c] = Bv[6 + c] + mem[j][0] * W[(6 + c) * 3 + 0] +
                    mem[j][1] * W[(6 + c) * 3 + 1] + mem[j][2] * W[(6 + c) * 3 + 2];
    }
  }
  __syncthreads();

  // exact incremental autoregressive decode, 63 steps
  for (int i = 0; i < 63; ++i) {
    if (j == 0) { yv[0] = tgt[i][0]; yv[1] = tgt[i][1]; yv[2] = tgt[i][2]; }
    __syncthreads();
    for (int l = 0; l < 4; ++l) {
      // ---- masked self-attention (KV cache) ----
      if (j == 0) {
        const float* W = sa_qkv_w + l * 27;
        const float* Bv = sa_qkv_b + l * 9;
        for (int c = 0; c < 3; ++c) {
          qv[c]        = Bv[c]     + yv[0]*W[c*3] + yv[1]*W[c*3+1] + yv[2]*W[c*3+2];
          ksa[l][i][c] = Bv[3 + c] + yv[0]*W[(3+c)*3] + yv[1]*W[(3+c)*3+1] + yv[2]*W[(3+c)*3+2];
          vsa[l][i][c] = Bv[6 + c] + yv[0]*W[(6+c)*3] + yv[1]*W[(6+c)*3+1] + yv[2]*W[(6+c)*3+2];
        }
      }
      __syncthreads();
      if (j <= i)
        sc[j] = (qv[0]*ksa[l][j][0] + qv[1]*ksa[l][j][1] + qv[2]*ksa[l][j][2]) * isq3;
      __syncthreads();
      if (j == 0) {
        float m = -1e30f;
        for (int k = 0; k <= i; ++k) m = fmaxf(m, sc[k]);
        float s = 0.f, o0 = 0.f, o1 = 0.f, o2 = 0.f;
        for (int k = 0; k <= i; ++k) {
          float e = __expf(sc[k] - m);
          s += e; o0 += e*vsa[l][k][0]; o1 += e*vsa[l][k][1]; o2 += e*vsa[l][k][2];
        }
        float inv = 1.f / s; o0 *= inv; o1 *= inv; o2 *= inv;
        const float* OW = sa_out_w + l * 9; const float* OB = sa_out_b + l * 3;
        float a0 = OB[0] + o0*OW[0] + o1*OW[1] + o2*OW[2];
        float a1 = OB[1] + o0*OW[3] + o1*OW[4] + o2*OW[5];
        float a2 = OB[2] + o0*OW[6] + o1*OW[7] + o2*OW[8];
        float v0 = yv[0]+a0, v1 = yv[1]+a1, v2 = yv[2]+a2;
        float mu = (v0+v1+v2)*(1.f/3.f);
        float d0 = v0-mu, d1 = v1-mu, d2 = v2-mu;
        float inv2 = rsqrtf((d0*d0+d1*d1+d2*d2)*(1.f/3.f) + 1e-5f);
        const float* LW = dln1_w + l*3; const float* LB = dln1_b + l*3;
        yv[0] = d0*inv2*LW[0]+LB[0]; yv[1] = d1*inv2*LW[1]+LB[1]; yv[2] = d2*inv2*LW[2]+LB[2];
      }
      __syncthreads();
      // ---- cross-attention over memory ----
      if (j == 0) {
        const float* W = ca_qkv_w + l * 27;
        const float* Bv = ca_qkv_b + l * 9;
        for (int c = 0; c < 3; ++c)
          qv[c] = Bv[c] + yv[0]*W[c*3] + yv[1]*W[c*3+1] + yv[2]*W[c*3+2];
      }
      __syncthreads();
      sc[j] = (qv[0]*kc[l][j][0] + qv[1]*kc[l][j][1] + qv[2]*kc[l][j][2]) * isq3;
      __syncthreads();
      if (j == 0) {
        float m = -1e30f;
        for (int k = 0; k < 64; ++k) m = fmaxf(m, sc[k]);
        float s = 0.f, o0 = 0.f, o1 = 0.f, o2 = 0.f;
        for (int k = 0; k < 64; ++k) {
          float e = __expf(sc[k] - m);
          s += e; o0 += e*vc[l][k][0]; o1 += e*vc[l][k][1]; o2 += e*vc[l][k][2];
        }
        float inv = 1.f / s; o0 *= inv; o1 *= inv; o2 *= inv;
        const float* OW = ca_out_w + l * 9; const float* OB = ca_out_b + l * 3;
        float a0 = OB[0] + o0*OW[0] + o1*OW[1] + o2*OW[2];
        float a1 = OB[1] + o0*OW[3] + o1*OW[4] + o2*OW[5];
        float a2 = OB[2] + o0*OW[6] + o1*OW[7] + o2*OW[8];
        float v0 = yv[0]+a0, v1 = yv[1]+a1, v2 = yv[2]+a2;
        float mu = (v0+v1+v2)*(1.f/3.f);
        float d0 = v0-mu, d1 = v1-mu, d2 = v2-mu;
        float inv2 = rsqrtf((d0*d0+d1*d1+d2*d2)*(1.f/3.f) + 1e-5f);
        const float* LW = dln2_w + l*3; const float* LB = dln2_b + l*3;
        yv[0] = d0*inv2*LW[0]+LB[0]; yv[1] = d1*inv2*LW[1]+LB[1]; yv[2] = d2*inv2*LW[2]+LB[2];

        // ---- feed-forward (3 -> 4 -> 3) + LN3, done in same thread ----
        const float* W1 = dlin1_w + l*12; const float* B1 = dlin1_b + l*4;
        const float* W2 = dlin2_w + l*12; const float* B2 = dlin2_b + l*3;
        float h[4];
        for (int f = 0; f < 4; ++f)
          h[f] = fmaxf(B1[f] + yv[0]*W1[f*3] + yv[1]*W1[f*3+1] + yv[2]*W1[f*3+2], 0.f);
        float z0 = B2[0] + h[0]*W2[0] + h[1]*W2[1] + h[2]*W2[2] + h[3]*W2[3];
        float z1 = B2[1] + h[0]*W2[4] + h[1]*W2[5] + h[2]*W2[6] + h[3]*W2[7];
        float z2 = B2[2] + h[0]*W2[8] + h[1]*W2[9] + h[2]*W2[10] + h[3]*W2[11];
        float u0 = yv[0]+z0, u1 = yv[1]+z1, u2 = yv[2]+z2;
        float mu3 = (u0+u1+u2)*(1.f/3.f);
        float e0 = u0-mu3, e1 = u1-mu3, e2 = u2-mu3;
        float inv3 = rsqrtf((e0*e0+e1*e1+e2*e2)*(1.f/3.f) + 1e-5f);
        const float* L3W = dln3_w + l*3; const float* L3B = dln3_b + l*3;
        yv[0] = e0*inv3*L3W[0]+L3B[0]; yv[1] = e1*inv3*L3W[1]+L3B[1]; yv[2] = e2*inv3*L3W[2]+L3B[2];
      }
      __syncthreads();
    }
    if (j == 0) { tgt[i+1][0] = yv[0]; tgt[i+1][1] = yv[1]; tgt[i+1][2] = yv[2]; }
    __syncthreads();
  }

  // output: [B,S,5] = concat(tgt, word idx, syll idx)
  float* op = out + (b * 64 + j) * 5;
  op[0] = tgt[j][0]; op[1] = tgt[j][1]; op[2] = tgt[j][2];
  op[3] = src[(b * 64 + j) * 16 + 14];
  op[4] = src[(b * 64 + j) * 16 + 15];
}

// ------------------------------ launch --------------------------------------
extern "C" void kernel_launch(void* const* d_in, const int* in_sizes, int n_in,
                              void* d_out, int out_size, void* d_ws, size_t ws_size,
                              hipStream_t stream) {
  (void)in_sizes; (void)n_in; (void)out_size; (void)ws_size;
  const float* src      = (const float*)d_in[0];
  const float* wemb     = (const float*)d_in[1];
  const float* semb     = (const float*)d_in[2];
  const float* eqkv_w   = (const float*)d_in[3];
  const float* eqkv_b   = (const float*)d_in[4];
  const float* eout_w   = (const float*)d_in[5];
  const float* eout_b   = (const float*)d_in[6];
  const float* eln1_w   = (const float*)d_in[7];
  const float* eln1_b   = (const float*)d_in[8];
  const float* elin1_w  = (const float*)d_in[9];
  const float* elin1_b  = (const float*)d_in[10];
  const float* elin2_w  = (const float*)d_in[11];
  const float* elin2_b  = (const float*)d_in[12];
  const float* eln2_w   = (const float*)d_in[13];
  const float* eln2_b   = (const float*)d_in[14];
  const float* norm_w   = (const float*)d_in[15];
  const float* norm_b   = (const float*)d_in[16];
  const float* eo_w     = (const float*)d_in[17];
  const float* eo_b     = (const float*)d_in[18];
  const float* saqkv_w  = (const float*)d_in[19];
  const float* saqkv_b  = (const float*)d_in[20];
  const float* saout_w  = (const float*)d_in[21];
  const float* saout_b  = (const float*)d_in[22];
  const float* dln1_w   = (const float*)d_in[23];
  const float* dln1_b   = (const float*)d_in[24];
  const float* caqkv_w  = (const float*)d_in[25];
  const float* caqkv_b  = (const float*)d_in[26];
  const float* caout_w  = (const float*)d_in[27];
  const float* caout_b  = (const float*)d_in[28];
  const float* dln2_w   = (const float*)d_in[29];
  const float* dln2_b   = (const float*)d_in[30];
  const float* dlin1_w  = (const float*)d_in[31];
  const float* dlin1_b  = (const float*)d_in[32];
  const float* dlin2_w  = (const float*)d_in[33];
  const float* dlin2_b  = (const float*)d_in[34];
  const float* dln3_w   = (const float*)d_in[35];
  const float* dln3_b   = (const float*)d_in[36];

  float* xg   = (float*)d_ws;                 // [64][64][36]
  float* memg = xg + 64 * 64 * 36;            // [64][64][3]

  embed_kernel<<<16, 256, 0, stream>>>(src, wemb, semb, xg);
  encoder_kernel<<<64, 256, 0, stream>>>(
      xg, eqkv_w, eqkv_b, eout_w, eout_b, eln1_w, eln1_b, elin1_w, elin1_b,
      elin2_w, elin2_b, eln2_w, eln2_b, norm_w, norm_b, eo_w, eo_b, memg);
  decoder_kernel<<<64, 64, 0, stream>>>(
      memg, src, saqkv_w, saqkv_b, saout_w, saout_b, dln1_w, dln1_b,
      caqkv_w, caqkv_b, caout_w, caout_b, dln2_w, dln2_b,
      dlin1_w, dlin1_b, dlin2_w, dlin2_b, dln3_w, dln3_b, (float*)d_out);
}